// VLLMFP8KVCache_72155450573434
// MI455X (gfx1250) — compile-verified
//
#include <hip/hip_runtime.h>
#include <stddef.h>

// ---- Problem constants (match reference) -----------------------------------
#define NUM_SLOTS  262144
#define H_HEADS    8
#define D_HEAD     128
#define ROW_ELEMS  (H_HEADS * D_HEAD)      // 1024 elements per slot row
#define NUM_TOKENS 8192
#define NUM_FETCH  32768
#define INPUT_SCALE  1.0f
#define OUTPUT_SCALE 1.0f

typedef __attribute__((ext_vector_type(2))) float        v2f;
typedef __attribute__((ext_vector_type(4))) unsigned int v4u;

// bf16 (raw 16-bit) -> f32: exact, just shift into the high half.
__device__ __forceinline__ float bf16_bits_to_f32(unsigned int b16) {
    union { unsigned int u; float f; } c;
    c.u = b16 << 16;
    return c.f;
}

// f32 -> bf16 raw bits, round-to-nearest-even (matches jnp astype(bfloat16)).
__device__ __forceinline__ unsigned int f32_to_bf16_bits(float f) {
    union { float f; unsigned int u; } c;
    c.f = f;
    unsigned int x = c.u;
    if ((x & 0x7fffffffu) > 0x7f800000u)            // NaN -> quiet NaN
        return ((x >> 16) | 0x0040u) & 0xffffu;
    unsigned int r = x + 0x7fffu + ((x >> 16) & 1u); // RNE
    return (r >> 16) & 0xffffu;
}

__device__ __forceinline__ unsigned int pack_bf16_pair(float lo, float hi) {
    return f32_to_bf16_bits(lo) | (f32_to_bf16_bits(hi) << 16);
}

// ---- Kernel 1: table[slot] = -1 for all slots ------------------------------
__global__ void kv_init_table(int* __restrict__ table) {
    // 4 ints per thread, b128 stores. NUM_SLOTS/4 v4u elements total.
    size_t i = (size_t)blockIdx.x * blockDim.x + threadIdx.x;
    v4u m1 = { 0xffffffffu, 0xffffffffu, 0xffffffffu, 0xffffffffu };
    ((v4u*)table)[i] = m1;
}

// ---- Kernel 2: scatter token ids: table[slot_mapping[t]] = t ---------------
__global__ void kv_scatter_map(const int* __restrict__ slot_mapping,
                               int* __restrict__ table) {
    int t = blockIdx.x * blockDim.x + threadIdx.x;
    table[slot_mapping[t]] = t;
}

// ---- Kernel 3: gather + (de)quantize ---------------------------------------
// 64 lanes per 1024-element row (16 elems / lane), 4 rows per 256-thread block.
__global__ void kv_gather(const unsigned short* __restrict__ inp,   // bf16 bits
                          const unsigned char*  __restrict__ cache, // fp8 e4m3
                          const int*            __restrict__ blocks,
                          const int*            __restrict__ table,
                          unsigned short*       __restrict__ out) { // bf16 bits
    const int lane     = threadIdx.x & 63;
    const int rowInBlk = threadIdx.x >> 6;
    const size_t row   = (size_t)blockIdx.x * 4 + rowInBlk;
    const int elem     = lane * 16;                 // element offset within row

    const int slot  = blocks[row];
    const int token = table[slot];

    float f[16];

    if (token < 0) {
        // Slot untouched this step: fp8 cache -> f32
        const v4u* src = (const v4u*)(cache + (size_t)slot * ROW_ELEMS + elem);
        v4u q = __builtin_nontemporal_load(src);
#pragma unroll
        for (int j = 0; j < 4; ++j) {
            unsigned int w = q[j];
            v2f lo = __builtin_amdgcn_cvt_pk_f32_fp8(w, false); // bytes 0,1
            v2f hi = __builtin_amdgcn_cvt_pk_f32_fp8(w, true);  // bytes 2,3
            f[4 * j + 0] = lo.x; f[4 * j + 1] = lo.y;
            f[4 * j + 2] = hi.x; f[4 * j + 3] = hi.y;
        }
    } else {
        // Slot overwritten this step: bf16 input -> fp8 round-trip -> f32
        const unsigned short* ip = inp + (size_t)token * ROW_ELEMS + elem;
        v4u a = __builtin_nontemporal_load((const v4u*)ip);
        v4u b = __builtin_nontemporal_load((const v4u*)ip + 1);
        unsigned int w8[8] = { a.x, a.y, a.z, a.w, b.x, b.y, b.z, b.w };
#pragma unroll
        for (int j = 0; j < 4; ++j) {
            float f0 = bf16_bits_to_f32(w8[2 * j]     & 0xffffu) * INPUT_SCALE;
            float f1 = bf16_bits_to_f32(w8[2 * j]     >> 16)     * INPUT_SCALE;
            float f2 = bf16_bits_to_f32(w8[2 * j + 1] & 0xffffu) * INPUT_SCALE;
            float f3 = bf16_bits_to_f32(w8[2 * j + 1] >> 16)     * INPUT_SCALE;
            // quantize: v_cvt_pk_fp8_f32 (2 elems/op)
            int packed = __builtin_amdgcn_cvt_pk_fp8_f32(f0, f1, 0,      false);
            packed     = __builtin_amdgcn_cvt_pk_fp8_f32(f2, f3, packed, true);
            // dequantize back: v_cvt_pk_f32_fp8
            v2f lo = __builtin_amdgcn_cvt_pk_f32_fp8((unsigned int)packed, false);
            v2f hi = __builtin_amdgcn_cvt_pk_f32_fp8((unsigned int)packed, true);
            f[4 * j + 0] = lo.x; f[4 * j + 1] = lo.y;
            f[4 * j + 2] = hi.x; f[4 * j + 3] = hi.y;
        }
    }

    // f32 -> bf16 out (two b128 non-temporal stores: 16 bf16 = 32 bytes)
    v4u o0, o1;
    o0.x = pack_bf16_pair(f[0]  * OUTPUT_SCALE, f[1]  * OUTPUT_SCALE);
    o0.y = pack_bf16_pair(f[2]  * OUTPUT_SCALE, f[3]  * OUTPUT_SCALE);
    o0.z = pack_bf16_pair(f[4]  * OUTPUT_SCALE, f[5]  * OUTPUT_SCALE);
    o0.w = pack_bf16_pair(f[6]  * OUTPUT_SCALE, f[7]  * OUTPUT_SCALE);
    o1.x = pack_bf16_pair(f[8]  * OUTPUT_SCALE, f[9]  * OUTPUT_SCALE);
    o1.y = pack_bf16_pair(f[10] * OUTPUT_SCALE, f[11] * OUTPUT_SCALE);
    o1.z = pack_bf16_pair(f[12] * OUTPUT_SCALE, f[13] * OUTPUT_SCALE);
    o1.w = pack_bf16_pair(f[14] * OUTPUT_SCALE, f[15] * OUTPUT_SCALE);

    v4u* op = (v4u*)(out + row * ROW_ELEMS + elem);
    __builtin_nontemporal_store(o0, op);
    __builtin_nontemporal_store(o1, op + 1);
}

// ---- Launcher ---------------------------------------------------------------
extern "C" void kernel_launch(void* const* d_in, const int* in_sizes, int n_in,
                              void* d_out, int out_size, void* d_ws, size_t ws_size,
                              hipStream_t stream) {
    const unsigned short* inp          = (const unsigned short*)d_in[0]; // bf16
    const unsigned char*  cache        = (const unsigned char*)d_in[1];  // fp8
    const int*            slot_mapping = (const int*)d_in[2];
    const int*            blocks       = (const int*)d_in[3];
    unsigned short*       out          = (unsigned short*)d_out;         // bf16
    int*                  table        = (int*)d_ws;                     // 1 MiB

    // 1) table = -1 everywhere (262144 ints, 4/thread)
    kv_init_table<<<NUM_SLOTS / (256 * 4), 256, 0, stream>>>(table);
    // 2) table[slot_mapping[t]] = t
    kv_scatter_map<<<NUM_TOKENS / 256, 256, 0, stream>>>(slot_mapping, table);
    // 3) gather + fp8 (de)quant; 4 rows per 256-thread block
    kv_gather<<<NUM_FETCH / 4, 256, 0, stream>>>(inp, cache, blocks, table, out);
}